// VectorQuantizer_33474975105238
// MI455X (gfx1250) — compile-verified
//
#include <hip/hip_runtime.h>
#include <hip/hip_bf16.h>
#include <math.h>

typedef __attribute__((ext_vector_type(2))) float v2f;
typedef __attribute__((ext_vector_type(8))) float v8f;
typedef __attribute__((ext_vector_type(4))) unsigned int u32x4;
typedef __attribute__((ext_vector_type(8))) int i32x8;
typedef __attribute__((ext_vector_type(4))) int i32x4;

#define DIM    256
#define NCODE  1024
#define NROWS  32768
#define ROWS_B 64            // rows per block (4 row-tiles of 16)
#define NBLK   (NROWS / ROWS_B)   // 512
#define XS     260           // padded LDS row stride (260 % 64 banks = 4 -> conflict-free column reads)

#define QOFF     ((size_t)1)
#define PERP_OFF ((size_t)8388609)
#define IDX_OFF  ((size_t)8388610)

// order-preserving (distance, index) -> u64 key; min(key) == argmin with smallest-index tie-break
__device__ __forceinline__ unsigned long long packKey(float v, int n) {
  unsigned int u = __float_as_uint(v);
  unsigned int s = u ^ (unsigned int)(((int)u >> 31) | 0x80000000);
  return ((unsigned long long)s << 32) | (unsigned int)n;
}

// ---------------- prep: transpose embedding, code norms, zero histogram ----------------
__global__ __launch_bounds__(256) void vq_prep(const float* __restrict__ E,
                                               float* __restrict__ ET,
                                               float* __restrict__ enorm,
                                               unsigned int* __restrict__ hist) {
  __shared__ float red[256];
  const int n = blockIdx.x;        // codebook entry 0..1023
  const int d = threadIdx.x;       // dim 0..255
  float v = E[(size_t)d * NCODE + n];
  ET[(size_t)n * DIM + d] = v;
  red[d] = v * v;
  __syncthreads();
  for (int s = 128; s > 0; s >>= 1) {
    if (d < s) red[d] += red[d + s];
    __syncthreads();
  }
  if (d == 0) { enorm[n] = red[0]; hist[n] = 0u; }
}

// ---------------- main: TDM tile load, fp32 WMMA distances, argmin, gather, SSE ----------------
__global__ __launch_bounds__(256) void vq_main(const float* __restrict__ x,
                                               const float* __restrict__ E,
                                               const float* __restrict__ ET,
                                               const float* __restrict__ enorm,
                                               unsigned int* __restrict__ hist,
                                               float* __restrict__ blockSse,
                                               float* __restrict__ out) {
  __shared__ float xt[ROWS_B * XS];
  __shared__ unsigned long long wBest[8][ROWS_B];
  __shared__ int   finalIdx[ROWS_B];
  __shared__ float red[256];

  const int tid    = threadIdx.x;
  const int wave   = tid >> 5;
  const int lane   = tid & 31;
  const int lane16 = lane & 15;
  const int koff   = (lane >> 4) * 2;     // A/B fragment K sub-offset per ISA layout
  const int m0     = blockIdx.x * ROWS_B;

  // ---- stage 64x256 f32 x-tile into LDS via Tensor Data Mover, with TDM padding
  // producing the 260-float row stride (pad 4 DWORDs every 256 DWORDs).
  if (wave == 0) {
    unsigned long long gaddr = (unsigned long long)(uintptr_t)(x + (size_t)m0 * DIM);
    unsigned int ldsoff = (unsigned int)(uintptr_t)&xt[0];   // low 32 bits = LDS byte offset
    u32x4 g0;
    g0[0] = 1u;                                              // count=1 (valid descriptor)
    g0[1] = ldsoff;                                          // lds_addr
    g0[2] = (unsigned int)(gaddr & 0xffffffffull);           // global_addr[31:0]
    g0[3] = (unsigned int)((gaddr >> 32) & 0x1ffffffull)     // global_addr[56:32]
          | (2u << 30);                                      // type=2 ("image")
    i32x8 g1;
    g1[0] = (int)(0x20000u          // data_size=2 -> 4-byte elements
          | (1u << 20)              // pad_enable
          | (7u << 22)              // pad_interval: 256 DWORDs
          | (3u << 25));            // pad_amount: 4 DWORDs  => 260-float LDS stride
    g1[1] = (int)((DIM & 0xffff) << 16);                     // tensor_dim0[15:0]=256
    g1[2] = (int)(((DIM >> 16) & 0xffff) | ((ROWS_B & 0xffff) << 16)); // dim0 hi | tensor_dim1=64
    g1[3] = (int)(((ROWS_B >> 16) & 0xffff) | (DIM << 16));  // dim1 hi | tile_dim0=256
    g1[4] = (int)(ROWS_B);                                   // tile_dim1=64, tile_dim2=0
    g1[5] = (int)DIM;                                        // tensor_dim0_stride=256
    g1[6] = 0;
    g1[7] = 0;
    i32x4 z4 = {0, 0, 0, 0};
    i32x8 z8 = {0, 0, 0, 0, 0, 0, 0, 0};
    __builtin_amdgcn_tensor_load_to_lds(g0, g1, z4, z4, z8, 0);
    __builtin_amdgcn_s_wait_tensorcnt(0);
  }
  __syncthreads();

  unsigned long long best[4][8];
#pragma unroll
  for (int rt = 0; rt < 4; ++rt)
#pragma unroll
    for (int r = 0; r < 8; ++r) best[rt][r] = 0xFFFFFFFFFFFFFFFFull;

  // each wave: 8 tiles of 16 codes; B fragment reused across 4 row-tiles
  for (int t = 0; t < 8; ++t) {
    const int n = wave * 128 + t * 16 + lane16;
    v8f cc[4];
#pragma unroll
    for (int rt = 0; rt < 4; ++rt) cc[rt] = (v8f){};
#pragma unroll 4
    for (int k = 0; k < DIM; k += 4) {
      v2f b;
      b.x = E[(size_t)(k + koff)     * NCODE + n];           // B: 4x16 fp32 fragment
      b.y = E[(size_t)(k + koff + 1) * NCODE + n];
#pragma unroll
      for (int rt = 0; rt < 4; ++rt) {
        v2f a = *(const v2f*)&xt[(rt * 16 + lane16) * XS + k + koff];  // A: 16x4 fragment
        cc[rt] = __builtin_amdgcn_wmma_f32_16x16x4_f32(false, a, false, b,
                                                       (short)0, cc[rt], false, false);
      }
    }
    float en = enorm[n];
#pragma unroll
    for (int rt = 0; rt < 4; ++rt)
#pragma unroll
      for (int r = 0; r < 8; ++r) {
        unsigned long long k = packKey(en - 2.0f * cc[rt][r], n);  // branchless running min
        best[rt][r] = (k < best[rt][r]) ? k : best[rt][r];
      }
  }

  // reduce within each 16-lane half (rows r for lanes 0-15, rows r+8 for lanes 16-31)
#pragma unroll
  for (int off = 8; off >= 1; off >>= 1) {
#pragma unroll
    for (int rt = 0; rt < 4; ++rt)
#pragma unroll
      for (int r = 0; r < 8; ++r) {
        unsigned long long ok =
            (unsigned long long)__shfl_xor((long long)best[rt][r], off, 32);
        best[rt][r] = (ok < best[rt][r]) ? ok : best[rt][r];
      }
  }
  if (lane16 == 0) {
    int half = (lane >> 4) * 8;
#pragma unroll
    for (int rt = 0; rt < 4; ++rt)
#pragma unroll
      for (int r = 0; r < 8; ++r)
        wBest[wave][rt * 16 + half + r] = best[rt][r];
  }
  __syncthreads();

  // final argmin across the 8 waves; emit indices + histogram
  if (tid < ROWS_B) {
    unsigned long long bk = wBest[0][tid];
    for (int w = 1; w < 8; ++w) {
      unsigned long long ok = wBest[w][tid];
      bk = (ok < bk) ? ok : bk;
    }
    int bi = (int)(bk & 0xFFFFFFFFull);
    finalIdx[tid] = bi;
    out[IDX_OFF + (size_t)(m0 + tid)] = (float)bi;
    atomicAdd(&hist[bi], 1u);
  }
  __syncthreads();

  // gather quantized rows (coalesced via ET) + per-block SSE
  float sse = 0.0f;
  for (int j = 0; j < ROWS_B; ++j) {
    int idx  = finalIdx[j];
    float q  = ET[(size_t)idx * DIM + tid];
    float xv = xt[j * XS + tid];
    float d  = q - xv;
    sse += d * d;
    out[QOFF + (size_t)(m0 + j) * DIM + tid] = q;
  }
  red[tid] = sse;
  __syncthreads();
  for (int s = 128; s > 0; s >>= 1) {
    if (tid < s) red[tid] += red[tid + s];
    __syncthreads();
  }
  if (tid == 0) blockSse[blockIdx.x] = red[0];
}

// ---------------- finalize: loss + perplexity (deterministic fixed-order sums) ----------------
__global__ __launch_bounds__(1024) void vq_final(const unsigned int* __restrict__ hist,
                                                 const float* __restrict__ blockSse,
                                                 float* __restrict__ out) {
  __shared__ float redP[1024];
  __shared__ float redS[1024];
  const int t = threadIdx.x;
  float p = (float)hist[t] * (1.0f / (float)NROWS);
  redP[t] = p * logf(p + 1e-10f);
  redS[t] = (t < NBLK) ? blockSse[t] : 0.0f;
  __syncthreads();
  for (int s = 512; s > 0; s >>= 1) {
    if (t < s) { redP[t] += redP[t + s]; redS[t] += redS[t + s]; }
    __syncthreads();
  }
  if (t == 0) {
    out[0]        = 1.25f * redS[0] / (float)((size_t)NROWS * DIM);  // q_loss + 0.25*e_loss
    out[PERP_OFF] = expf(-redP[0]);
  }
}

extern "C" void kernel_launch(void* const* d_in, const int* in_sizes, int n_in,
                              void* d_out, int out_size, void* d_ws, size_t ws_size,
                              hipStream_t stream) {
  const float* x = (const float*)d_in[0];   // [32,32,32,256]
  const float* E = (const float*)d_in[1];   // [256,1024]
  float* out = (float*)d_out;

  float* ws            = (float*)d_ws;
  float* ET            = ws;                              // 262144 floats (1 MB)
  float* enorm         = ws + 262144;                     // 1024 floats
  unsigned int* hist   = (unsigned int*)(ws + 263168);    // 1024 u32
  float* blockSse      = ws + 264192;                     // 512 floats

  vq_prep <<<NCODE, 256, 0, stream>>>(E, ET, enorm, hist);
  vq_main <<<NBLK, 256, 0, stream>>>(x, E, ET, enorm, hist, blockSse, out);
  vq_final<<<1, 1024, 0, stream>>>(hist, blockSse, out);
}